// Interaction_28776280884018
// MI455X (gfx1250) — compile-verified
//
#include <hip/hip_runtime.h>

typedef __attribute__((ext_vector_type(2))) float v2f;
typedef __attribute__((ext_vector_type(4))) float v4f;
typedef __attribute__((ext_vector_type(8))) float v8f;

#define BDIM 32
#define NDIM 16
#define LDIM 64
#define DDIM 63
#define DP   128
#define LCHUNK 16
#define NLC  (LDIM / LCHUNK)   // 4 l-chunks

// out[b, i*16+j, k, l] = sum_m s[b,(j-i)%16,l,m] * t[b,j,l,k-m]
// One workgroup per (b, j, l-chunk). 256 threads = 8 waves, each wave owns one l per round.
__global__ __launch_bounds__(256) void Interaction_conv_wmma_kernel(
    const float* __restrict__ s, const float* __restrict__ t, float* __restrict__ out)
{
    __shared__ float s_stage[NDIM][LCHUNK][64];  // 64 KB  (m padded 63->64 with 0)
    __shared__ float t_stage[LCHUNK][256];       // 16 KB  (index shift +64, zero-padded both sides)
    __shared__ float o_stage[NDIM][DP][8];       // 64 KB  (a, k, l-slice of 8)

    const int tid  = threadIdx.x;
    const int lane = tid & 31;
    const int w    = tid >> 5;      // wave id 0..7
    const int hi   = lane >> 4;     // half-wave select
    const int lo16 = lane & 15;

    const int wg = blockIdx.x;      // 0..2047
    const int b  = wg >> 6;
    const int j  = (wg >> 2) & 15;
    const int lc = wg & 3;
    const int l0 = lc * LCHUNK;

    // ---- stage S[b, :, l0..l0+15, 0..63] (zero pad at m=63), coalesced over m ----
    for (int it = 0; it < 64; ++it) {
        int idx = tid + it * 256;            // 16*16*64 = 16384 elements
        int m   = idx & 63;
        int r   = idx >> 6;
        int ll  = r & 15;
        int a   = r >> 4;
        float v = 0.f;
        if (m < DDIM)
            v = s[(((size_t)b * NDIM + a) * LDIM + (l0 + ll)) * DDIM + m];
        s_stage[a][ll][m] = v;
    }
    // ---- stage T[b, j, l0..l0+15, :] with 64-zero guard bands: t_stage[ll][v+64] ----
    for (int it = 0; it < 16; ++it) {
        int idx = tid + it * 256;            // 16*256 = 4096 elements
        int q   = idx & 255;
        int ll  = idx >> 8;
        int vi  = q - 64;
        float v = 0.f;
        if (vi >= 0 && vi < DDIM)
            v = t[(((size_t)b * NDIM + j) * LDIM + (l0 + ll)) * DDIM + vi];
        t_stage[ll][q] = v;
    }
    __syncthreads();

    for (int r2 = 0; r2 < 2; ++r2) {
        const int ll = r2 * 8 + w;           // this wave's l within the chunk

        // A fragments (16x4 f32 layout): lane holds row a=lo16, K = 4*kk + 2*hi + {0,1}
        v2f afrag[16];
        #pragma unroll
        for (int kk = 0; kk < 16; ++kk) {
            int mbase = 4 * kk + 2 * hi;
            afrag[kk] = *(const v2f*)&s_stage[lo16][ll][mbase];
        }

        #pragma unroll
        for (int nt = 0; nt < 8; ++nt) {
            const int kcol = nt * 16 + lo16; // B column / D column this lane owns
            v8f acc0 = {};
            v8f acc1 = {};
            // Inline Toeplitz B loads: GVN shares t-values across (nt,kk) tiles
            // (index 16*nt - 4*kk collides for nt+1,kk+4), 2 accumulator chains.
            #pragma unroll
            for (int kk = 0; kk < 16; kk += 2) {
                {   // B (4x16 f32 layout): lane holds col kcol, K = 4*kk + 2*hi + {0,1}
                    int mbase = 4 * kk + 2 * hi;
                    int q0 = kcol - mbase + 64;      // shift-padded index, always in range
                    v2f bfrag;
                    bfrag.x = t_stage[ll][q0];       // t[kcol - mbase]
                    bfrag.y = t_stage[ll][q0 - 1];   // t[kcol - mbase - 1]
                    acc0 = __builtin_amdgcn_wmma_f32_16x16x4_f32(
                        false, afrag[kk], false, bfrag, (short)0, acc0, false, false);
                }
                {
                    int mbase = 4 * (kk + 1) + 2 * hi;
                    int q0 = kcol - mbase + 64;
                    v2f bfrag;
                    bfrag.x = t_stage[ll][q0];
                    bfrag.y = t_stage[ll][q0 - 1];
                    acc1 = __builtin_amdgcn_wmma_f32_16x16x4_f32(
                        false, afrag[kk + 1], false, bfrag, (short)0, acc1, false, false);
                }
            }
            // D layout: VGPR r -> (a = r + 8*hi, n = lo16)
            #pragma unroll
            for (int r = 0; r < 8; ++r) {
                int arow = r + 8 * hi;
                o_stage[arow][kcol][w] = acc0[r] + acc1[r];
            }
        }
        __syncthreads();

        // ---- cooperative write: o_stage[a][k][0..7] -> out[b, p(a), k, l0 + r2*8 + 0..7] ----
        #pragma unroll
        for (int it = 0; it < 8; ++it) {
            int row = tid + it * 256;        // 2048 (a,k) rows
            int a = row >> 7;
            int k = row & 127;
            int i = (j - a) & 15;
            int p = i * 16 + j;
            size_t go = (((size_t)b * 256 + p) * DP + k) * (size_t)LDIM + l0 + r2 * 8;
            const float* srcp = &o_stage[a][k][0];
            *(v4f*)&out[go]     = *(const v4f*)&srcp[0];
            *(v4f*)&out[go + 4] = *(const v4f*)&srcp[4];
        }
        __syncthreads();
    }
}

extern "C" void kernel_launch(void* const* d_in, const int* in_sizes, int n_in,
                              void* d_out, int out_size, void* d_ws, size_t ws_size,
                              hipStream_t stream) {
    (void)in_sizes; (void)n_in; (void)out_size; (void)d_ws; (void)ws_size;
    const float* s = (const float*)d_in[0];
    const float* t = (const float*)d_in[1];
    float* out = (float*)d_out;
    dim3 grid(BDIM * NDIM * NLC);   // 32 * 16 * 4 = 2048 workgroups
    Interaction_conv_wmma_kernel<<<grid, 256, 0, stream>>>(s, t, out);
}